// GraphConvolution_70403103916520
// MI455X (gfx1250) — compile-verified
//
#include <hip/hip_runtime.h>

typedef __attribute__((ext_vector_type(2))) float v2f;
typedef __attribute__((ext_vector_type(4))) float v4f;
typedef __attribute__((ext_vector_type(8))) float v8f;

#define N_NODES 50000
#define KNBR    32
#define DIN     256
#define DOUT    256
#define M_TILE  32          // nodes per block
#define LDS_STRIDE 260      // 256 + 4 pad -> conflict-free column reads

__global__ __launch_bounds__(256) void gconv_fused_kernel(
    const float* __restrict__ feats,    // [N, 256]
    const float* __restrict__ W,        // [256, 256] row-major: W[o][d]
    const float* __restrict__ bias,     // [256]
    const int*   __restrict__ nbrs,     // [N, 32]
    float*       __restrict__ out)      // [N, 256]
{
    __shared__ float sA[M_TILE * LDS_STRIDE];   // pooled tile, ~33 KB

    const int tid       = threadIdx.x;
    const int lane      = tid & 31;
    const int wave      = tid >> 5;             // 0..7
    const int blockBase = blockIdx.x * M_TILE;

    // ---------------- Stage 1: gather + mean-pool 32 nodes into LDS ----------
    // wave w pools nodes [4w, 4w+4); each lane owns 8 contiguous dims.
    for (int mm = wave * 4; mm < wave * 4 + 4; ++mm) {
        const int n = blockBase + mm;           // wave-uniform
        v4f acc0 = {0.f, 0.f, 0.f, 0.f};
        v4f acc1 = {0.f, 0.f, 0.f, 0.f};
        if (n < N_NODES) {
            const int* nb = nbrs + (size_t)n * KNBR;
            #pragma unroll 4
            for (int k = 0; k < KNBR; ++k) {
                const float* row = feats + (size_t)nb[k] * DIN + lane * 8;
                acc0 += *(const v4f*)(row);
                acc1 += *(const v4f*)(row + 4);
            }
            acc0 *= 0.03125f;   // 1/32 mean
            acc1 *= 0.03125f;
        }
        // invalid tail nodes store zeros so WMMA reads defined data (EXEC all-1s later)
        float* dst = &sA[mm * LDS_STRIDE + lane * 8];
        *(v4f*)(dst)     = acc0;
        *(v4f*)(dst + 4) = acc1;
    }
    __syncthreads();

    // ---------------- Stage 2: [32x256] x [256x256]^T via V_WMMA_F32_16X16X4_F32
    // wave -> one 16-row M-tile (wave&1) and four 16-col N-tiles ((wave>>1)*64..+63)
    const int m_tile  = wave & 1;
    const int n_group = wave >> 1;
    const int row     = lane & 15;
    const int koff    = (lane >> 4) * 2;        // lanes 16-31 carry K+2,K+3 (ISA A layout)

    const float* aBase = &sA[(m_tile * 16 + row) * LDS_STRIDE + koff];
    const float* wBase = W + (size_t)(n_group * 64 + row) * DIN + koff;

    v8f acc[4];
    {
        const v8f zero = {0.f, 0.f, 0.f, 0.f, 0.f, 0.f, 0.f, 0.f};
        #pragma unroll
        for (int j = 0; j < 4; ++j) acc[j] = zero;
    }

    #pragma unroll 4
    for (int kk = 0; kk < DIN; kk += 4) {
        // A fragment: pooled[m_tile*16+row][kk+koff .. +1]  (8B-aligned LDS read)
        const v2f a = *(const v2f*)(aBase + kk);
        #pragma unroll
        for (int j = 0; j < 4; ++j) {
            // B fragment: B[k][o] = W[o][k]; N (=o) across lanes, K pair per lane
            const v2f bf = *(const v2f*)(wBase + (size_t)j * 16 * DIN + kk);
            acc[j] = __builtin_amdgcn_wmma_f32_16x16x4_f32(
                /*neg_a=*/false, a, /*neg_b=*/false, bf,
                /*c_mod=*/(short)0, acc[j],
                /*reuse_a=*/false, /*reuse_b=*/false);
        }
    }

    // ---------------- Stage 3: bias + ReLU epilogue -------------------------
    // C/D layout: VGPR r -> M = r (lanes 0-15) or r+8 (lanes 16-31); N = lane&15
    const int  mrow0     = m_tile * 16 + ((lane >> 4) * 8);
    const bool fullBlock = (blockBase + M_TILE) <= N_NODES;   // uniform: all but last block

    if (fullBlock) {
        #pragma unroll
        for (int j = 0; j < 4; ++j) {
            const int   o  = n_group * 64 + j * 16 + row;
            const float bo = bias[o];
            float* outBase = out + (size_t)(blockBase + mrow0) * DOUT + o;
            #pragma unroll
            for (int r = 0; r < 8; ++r) {
                const float v = acc[j][r] + bo;
                outBase[(size_t)r * DOUT] = v > 0.f ? v : 0.f;
            }
        }
    } else {
        #pragma unroll
        for (int j = 0; j < 4; ++j) {
            const int   o  = n_group * 64 + j * 16 + row;
            const float bo = bias[o];
            #pragma unroll
            for (int r = 0; r < 8; ++r) {
                const int n = blockBase + mrow0 + r;
                if (n < N_NODES) {
                    const float v = acc[j][r] + bo;
                    out[(size_t)n * DOUT + o] = v > 0.f ? v : 0.f;
                }
            }
        }
    }
}

extern "C" void kernel_launch(void* const* d_in, const int* in_sizes, int n_in,
                              void* d_out, int out_size, void* d_ws, size_t ws_size,
                              hipStream_t stream) {
    (void)in_sizes; (void)n_in; (void)out_size; (void)d_ws; (void)ws_size;
    const float* feats = (const float*)d_in[0];   // [50000, 256] f32
    const float* W     = (const float*)d_in[1];   // [256, 256]   f32
    const float* bias  = (const float*)d_in[2];   // [256]        f32
    const int*   nbrs  = (const int*)  d_in[3];   // [50000, 32]  int32 (JAX default)
    float*       out   = (float*)d_out;           // [50000, 256] f32

    const int nblocks = (N_NODES + M_TILE - 1) / M_TILE;   // 1563
    gconv_fused_kernel<<<nblocks, 256, 0, stream>>>(feats, W, bias, nbrs, out);
}